// YOLOPredict_62569083568620
// MI455X (gfx1250) — compile-verified
//
#include <hip/hip_runtime.h>
#include <hip/hip_bf16.h>

#define S_GRID 32
#define A_NUM 5
#define C_CLS 20
#define N_BOX 5120          // S*S*A
#define WORDS 160           // N/32 bitmask words per row
#define CONF_T 0.1f
#define PAD 8192            // next pow2 >= N for bitonic sort

typedef float v2f __attribute__((ext_vector_type(2)));
typedef float v8f __attribute__((ext_vector_type(8)));

// ---------------------------------------------------------------------------
// Kernel 1: decode boxes (center->corner, clipped to unit image) + areas
// ---------------------------------------------------------------------------
__global__ void yolo_decode(const float* __restrict__ bbox,
                            const float* __restrict__ anchors,
                            float4* __restrict__ boxes,
                            float* __restrict__ area) {
    int i = blockIdx.x * blockDim.x + threadIdx.x;
    if (i >= N_BOX) return;
    int a  = i % A_NUM;
    int gj = (i / A_NUM) % S_GRID;
    int gi = i / (A_NUM * S_GRID);
    const float* bp = bbox + (size_t)i * 4;
    float cx = (bp[0] + (float)gi) * (1.0f / S_GRID);
    float cy = (bp[1] + (float)gj) * (1.0f / S_GRID);
    float w  = bp[2] * (anchors[a * 2 + 0] * (1.0f / 17.0f)) * (1.0f / S_GRID);
    float h  = bp[3] * (anchors[a * 2 + 1] * (1.0f / 17.0f)) * (1.0f / S_GRID);
    float x1 = fminf(fmaxf(cx - 0.5f * w, 0.0f), 1.0f);
    float y1 = fminf(fmaxf(cy - 0.5f * h, 0.0f), 1.0f);
    float x2 = fminf(fmaxf(cx + 0.5f * w, 0.0f), 1.0f);
    float y2 = fminf(fmaxf(cy + 0.5f * h, 0.0f), 1.0f);
    boxes[i] = make_float4(x1, y1, x2, y2);
    area[i]  = (x2 - x1) * (y2 - y1);
}

// ---------------------------------------------------------------------------
// Kernel 2: (iou > 0.5) bit-matrix. One wave32 per 16-row x 32-col block.
// union = area_i + area_j - inter computed by V_WMMA_F32_16X16X4_F32 with the
// rank-2 trick: A=[area_row,1,0,0] (16x4), B=[1; area_col; 0; 0] (4x16),
// C = -inter. Predicate packed via wave32 ballot (C/D layout: ballot bits
// [15:0] = matrix row v, bits [31:16] = row v+8).
// ---------------------------------------------------------------------------
__global__ void __launch_bounds__(32)
yolo_ioumask(const float4* __restrict__ boxes,
             const float* __restrict__ area,
             unsigned* __restrict__ supmask) {
    const int lane    = threadIdx.x;           // 0..31
    const int word    = blockIdx.x;            // 0..159 (32 cols each)
    const int rowBase = blockIdx.y * 16;       // 0..5104
    const int n       = lane & 15;
    const int hi      = lane >> 4;             // 0: rows v, 1: rows v+8

    // 8 row boxes this lane needs (C/D layout: M = v + 8*hi)
    float4 rb[8];
#pragma unroll
    for (int v = 0; v < 8; ++v) rb[v] = boxes[rowBase + hi * 8 + v];

    // A matrix 16x4: vgpr0 = K0 (area_row), vgpr1 = K1 (ones); K2=K3=0
    int aidx = (lane < 16) ? (rowBase + lane) : rowBase;
    float arow = area[aidx];
    v2f amat;
    amat.x = (lane < 16) ? arow : 0.0f;
    amat.y = (lane < 16) ? 1.0f : 0.0f;

    unsigned rowbits[16];
#pragma unroll
    for (int r = 0; r < 16; ++r) rowbits[r] = 0u;

#pragma unroll
    for (int cT = 0; cT < 2; ++cT) {
        int col = word * 32 + cT * 16 + n;
        float4 cb = boxes[col];
        int bidx = word * 32 + cT * 16 + ((lane < 16) ? lane : 0);
        float acol = area[bidx];

        // B matrix 4x16: K0 row = ones, K1 row = area_col; K2=K3=0
        v2f bmat;
        bmat.x = (lane < 16) ? 1.0f : 0.0f;
        bmat.y = (lane < 16) ? acol : 0.0f;

        // intersection per C/D element, negated into the accumulator
        v8f c;
        float inter[8];
#pragma unroll
        for (int v = 0; v < 8; ++v) {
            float4 r = rb[v];
            float lx = fmaxf(r.x, cb.x);
            float ly = fmaxf(r.y, cb.y);
            float rx = fminf(r.z, cb.z);
            float ry = fminf(r.w, cb.w);
            float iw = fmaxf(rx - lx, 0.0f);
            float ih = fmaxf(ry - ly, 0.0f);
            inter[v] = iw * ih;
            c[v] = -inter[v];
        }

        // union = area_i + area_j - inter  (matrix pipe does the rank-2 sums)
        v8f u = __builtin_amdgcn_wmma_f32_16x16x4_f32(
            false, amat, false, bmat, (short)0, c, false, false);

#pragma unroll
        for (int v = 0; v < 8; ++v) {
            float un = u[v];
            bool pred = (un > 0.0f) && (inter[v] > 0.5f * un);
            unsigned b32 = (unsigned)__ballot(pred);
            rowbits[v]     |= (b32 & 0xFFFFu) << (16 * cT);
            rowbits[v + 8] |= (b32 >> 16)     << (16 * cT);
        }
    }

    if (lane == 0) {
#pragma unroll
        for (int r = 0; r < 16; ++r)
            supmask[(size_t)(rowBase + r) * WORDS + word] = rowbits[r];
    }
}

// ---------------------------------------------------------------------------
// Kernel 3: per-class sort (LDS bitonic) + sequential wave32 NMS over the
// bit-matrix + masked output. One 1024-thread workgroup per class.
// ---------------------------------------------------------------------------
__device__ __forceinline__ bool sort_before(float ka, unsigned short ia,
                                            float kb, unsigned short ib) {
    // score descending, index ascending on ties (matches stable argsort(-sc))
    return (ka > kb) || (ka == kb && ia < ib);
}

__global__ void __launch_bounds__(1024)
yolo_nms(const float4* __restrict__ boxes,
         const float* __restrict__ conf,
         const float* __restrict__ score,
         const unsigned* __restrict__ supmask,
         float* __restrict__ out) {
    __shared__ float          s_key[PAD];
    __shared__ unsigned short s_idx[PAD];
    __shared__ unsigned       s_sup[WORDS];
    __shared__ unsigned char  s_keep[N_BOX];

    const int cls = blockIdx.x;
    const float NEG_INF = -__builtin_inff();

    // fill keys (invalid -> -inf; pad entries sort after real invalids by idx)
    for (int i = threadIdx.x; i < PAD; i += blockDim.x) {
        float key = NEG_INF;
        if (i < N_BOX) {
            float sc0 = score[(size_t)i * C_CLS + cls];
            if (sc0 > CONF_T) key = conf[i] * sc0;
        }
        s_key[i] = key;
        s_idx[i] = (unsigned short)i;
    }
    if (threadIdx.x < WORDS) s_sup[threadIdx.x] = 0u;

    // bitonic sort of 8192 (key desc, idx asc)
    for (unsigned k = 2; k <= PAD; k <<= 1) {
        for (unsigned j = k >> 1; j > 0; j >>= 1) {
            __syncthreads();
            for (unsigned t = threadIdx.x; t < PAD; t += blockDim.x) {
                unsigned ixj = t ^ j;
                if (ixj > t) {
                    float k0 = s_key[t], k1 = s_key[ixj];
                    unsigned short i0 = s_idx[t], i1 = s_idx[ixj];
                    bool up = ((t & k) == 0);
                    bool sw = up ? sort_before(k1, i1, k0, i0)
                                 : sort_before(k0, i0, k1, i1);
                    if (sw) {
                        s_key[t] = k1; s_key[ixj] = k0;
                        s_idx[t] = i1; s_idx[ixj] = i0;
                    }
                }
            }
        }
    }
    __syncthreads();

    // sequential greedy NMS: one wave32 owns the suppression bitset
    if (threadIdx.x < 32) {
        const int lane = threadIdx.x;
        volatile unsigned* vsup = s_sup;
        for (int t = 0; t < N_BOX; ++t) {
            int i = (int)s_idx[t];
            bool valid = (s_key[t] != NEG_INF);
            unsigned supw = vsup[i >> 5];            // uniform broadcast read
            bool suppressed = (supw >> (i & 31)) & 1u;
            bool kept = valid && !suppressed;
            if (lane == 0) s_keep[t] = kept ? 1 : 0;
            if (kept) {
                const unsigned* row = supmask + (size_t)i * WORDS;
                for (int w = lane; w < WORDS; w += 32)
                    vsup[w] |= row[w];               // 5 words per lane
            }
        }
    }
    __syncthreads();

    // masked output: [cls, slot, {x1,y1,x2,y2,score}]
    for (int t = threadIdx.x; t < N_BOX; t += blockDim.x) {
        int i = (int)s_idx[t];
        float m = s_keep[t] ? 1.0f : 0.0f;
        float4 b = boxes[i];
        float sc = conf[i] * score[(size_t)i * C_CLS + cls];
        float* o = out + ((size_t)cls * N_BOX + t) * 5;
        o[0] = b.x * m;
        o[1] = b.y * m;
        o[2] = b.z * m;
        o[3] = b.w * m;
        o[4] = sc * m;
    }
}

// ---------------------------------------------------------------------------
extern "C" void kernel_launch(void* const* d_in, const int* in_sizes, int n_in,
                              void* d_out, int out_size, void* d_ws, size_t ws_size,
                              hipStream_t stream) {
    const float* bbox    = (const float*)d_in[0];  // [S,S,A,4]
    const float* ioup    = (const float*)d_in[1];  // [N,1]
    const float* scorep  = (const float*)d_in[2];  // [N,C]
    const float* anchors = (const float*)d_in[3];  // [A,2]
    float* out = (float*)d_out;                    // [C,N,5]

    char* w = (char*)d_ws;
    float4*   boxes = (float4*)w;                                 // 81920 B
    float*    area  = (float*)(w + (size_t)N_BOX * 16);           // 20480 B
    unsigned* sup   = (unsigned*)(w + (size_t)N_BOX * 16
                                    + (size_t)N_BOX * 4);         // 3.28 MB

    yolo_decode<<<(N_BOX + 255) / 256, 256, 0, stream>>>(bbox, anchors, boxes, area);

    dim3 gmask(WORDS, N_BOX / 16);
    yolo_ioumask<<<gmask, 32, 0, stream>>>(boxes, area, sup);

    yolo_nms<<<C_CLS, 1024, 0, stream>>>(boxes, ioup, scorep, sup, out);
}